// NonLocalBlock_21242908246190
// MI455X (gfx1250) — compile-verified
//
#include <hip/hip_runtime.h>

// NonLocal block, MI455X (gfx1250, wave32). Flash-style: scores [4,4096,4096]
// (256MB) are never materialized; both GEMMs run on V_WMMA_F32_16X16X4_F32.
// Softmax is computed in the log2 domain (log2e folded into delta at
// projection time) so every exp is a bare v_exp_f32 with no scaling multiply.

typedef float v2f __attribute__((ext_vector_type(2)));
typedef float v8f __attribute__((ext_vector_type(8)));

#define B_N 4
#define C_N 64
#define C1_N 8
#define C2_N 32
#define HW_N 4096
#define RT_PER_WG 8   // 8 waves per WG, one 16-row tile each -> 128 rows per WG
#define WG_PER_B 32   // 32 WGs * 128 rows = 4096 rows per batch

#define LOG2E 1.4426950408889634f

__device__ __forceinline__ float exp2f_hw(float x) {
  return __builtin_amdgcn_exp2f(x);       // bare v_exp_f32
}

__device__ __forceinline__ v8f wmma4(v2f a, v2f b, v8f c) {
  // D(16x16,f32) = A(16x4,f32) * B(4x16,f32) + C
  return __builtin_amdgcn_wmma_f32_16x16x4_f32(
      /*neg_a=*/false, a, /*neg_b=*/false, b,
      /*c_mod=*/(short)0, c, /*reuse_a=*/false, /*reuse_b=*/false);
}

// ---------------------------------------------------------------------------
// Pass 1: projections. delta[b][n][8] (pre-scaled by log2e), phi[(c1*4+b)][n]
// (reference's transpose(1,0,2,3).reshape remix), g[b][n][32].
// ---------------------------------------------------------------------------
__global__ void __launch_bounds__(256)
proj_kernel(const float* __restrict__ x,
            const float* __restrict__ Wd, const float* __restrict__ bd,
            const float* __restrict__ Wp, const float* __restrict__ bp,
            const float* __restrict__ Wg, const float* __restrict__ bg,
            float* __restrict__ delta, float* __restrict__ phi,
            float* __restrict__ g) {
  int tid = blockIdx.x * blockDim.x + threadIdx.x;
  int b = tid >> 12;
  int n = tid & (HW_N - 1);
  float dacc[C1_N], pacc[C1_N], gacc[C2_N];
#pragma unroll
  for (int j = 0; j < C1_N; ++j) { dacc[j] = 0.f; pacc[j] = 0.f; }
#pragma unroll
  for (int j = 0; j < C2_N; ++j) gacc[j] = 0.f;
  const float* xb = x + (b * C_N) * HW_N + n;
  for (int c = 0; c < C_N; ++c) {
    float xv = xb[c * HW_N];
#pragma unroll
    for (int j = 0; j < C1_N; ++j) {
      dacc[j] = fmaf(xv, Wd[j * C_N + c], dacc[j]);
      pacc[j] = fmaf(xv, Wp[j * C_N + c], pacc[j]);
    }
#pragma unroll
    for (int j = 0; j < C2_N; ++j)
      gacc[j] = fmaf(xv, Wg[j * C_N + c], gacc[j]);
  }
#pragma unroll
  for (int j = 0; j < C1_N; ++j) {
    // scores feed exp2 later: fold log2(e) into delta (its only consumer).
    delta[(b * HW_N + n) * C1_N + j] = (dacc[j] + bd[j]) * LOG2E;
    int remix = j * B_N + b;                 // transpose(1,0,2,3).reshape quirk
    phi[remix * HW_N + n] = pacc[j] + bp[j];
  }
#pragma unroll
  for (int j = 0; j < C2_N; ++j)
    g[(b * HW_N + n) * C2_N + j] = gacc[j] + bg[j];
}

// ---------------------------------------------------------------------------
// Pass 2: online (max, sum-exp2) of log2-domain scores per workgroup.
// S^T tile via WMMA: A = phi^T (M=m, K=c1), B = delta^T (K=c1, N=r), K=8=2xK4.
// ---------------------------------------------------------------------------
__global__ void __launch_bounds__(256)
stats_kernel(const float* __restrict__ delta, const float* __restrict__ phi,
             float* __restrict__ maxp, float* __restrict__ sump) {
  __shared__ float wmax[RT_PER_WG], wsum[RT_PER_WG];
  int wg = blockIdx.x;
  int b = wg / WG_PER_B;
  int rblk = wg % WG_PER_B;
  int wave = threadIdx.x >> 5;
  int lane = threadIdx.x & 31;
  int half = lane >> 4;
  int l16 = lane & 15;
  int r0 = (rblk * RT_PER_WG + wave) * 16;

  v2f db[2];                                  // delta^T B-operand (loop-invariant)
#pragma unroll
  for (int kc = 0; kc < 2; ++kc) {
    int k = kc * 4 + half * 2;
    const float* dp = delta + (b * HW_N + r0 + l16) * C1_N + k;
    db[kc].x = dp[0];
    db[kc].y = dp[1];
  }

  float mx = -__builtin_inff(), sm = 0.f;
  const float* phib = phi + b * C1_N * HW_N;
  for (int mt = 0; mt < HW_N / 16; ++mt) {
    v2f a[2];
#pragma unroll
    for (int kc = 0; kc < 2; ++kc) {
      int k = kc * 4 + half * 2;
      const float* pp = phib + k * HW_N + mt * 16 + l16;
      a[kc].x = pp[0];
      a[kc].y = pp[HW_N];
    }
    v8f s = {0.f, 0.f, 0.f, 0.f, 0.f, 0.f, 0.f, 0.f};
    s = wmma4(a[0], db[0], s);
    s = wmma4(a[1], db[1], s);
    float tmax = s[0];
#pragma unroll
    for (int i = 1; i < 8; ++i) tmax = fmaxf(tmax, s[i]);
    float nm = fmaxf(mx, tmax);
    sm *= exp2f_hw(mx - nm);
#pragma unroll
    for (int i = 0; i < 8; ++i) sm += exp2f_hw(s[i] - nm);
    mx = nm;
  }
#pragma unroll
  for (int off = 16; off >= 1; off >>= 1) {    // lane reduce (wave32)
    float om = __shfl_xor(mx, off, 32);
    float os = __shfl_xor(sm, off, 32);
    float nm = fmaxf(mx, om);
    sm = sm * exp2f_hw(mx - nm) + os * exp2f_hw(om - nm);
    mx = nm;
  }
  if (lane == 0) { wmax[wave] = mx; wsum[wave] = sm; }
  __syncthreads();
  if (threadIdx.x == 0) {
    float M = wmax[0], S = wsum[0];
    for (int i = 1; i < RT_PER_WG; ++i) {
      float nm = fmaxf(M, wmax[i]);
      S = S * exp2f_hw(M - nm) + wsum[i] * exp2f_hw(wmax[i] - nm);
      M = nm;
    }
    maxp[b * WG_PER_B + rblk] = M;
    sump[b * WG_PER_B + rblk] = S;
  }
}

__global__ void reduce_stats_kernel(const float* __restrict__ maxp,
                                    const float* __restrict__ sump,
                                    float* __restrict__ Mb, float* __restrict__ Sb) {
  int b = threadIdx.x;
  if (b < B_N) {
    float M = -__builtin_inff(), S = 0.f;
    for (int i = 0; i < WG_PER_B; ++i) {
      float m = maxp[b * WG_PER_B + i], s = sump[b * WG_PER_B + i];
      float nm = fmaxf(M, m);
      S = S * exp2f_hw(M - nm) + s * exp2f_hw(m - nm);
      M = nm;
    }
    Mb[b] = M;
    Sb[b] = S;
  }
}

// ---------------------------------------------------------------------------
// Pass 3: recompute S^T tile -> exp2 -> half-wave shuffle into B-operand
// layout -> out^T += g^T * P^T (WMMA) -> scale 1/S -> fused conv + residual.
// ---------------------------------------------------------------------------
__global__ void __launch_bounds__(256)
out_kernel(const float* __restrict__ x, const float* __restrict__ delta,
           const float* __restrict__ phi, const float* __restrict__ g,
           const float* __restrict__ Wl, const float* __restrict__ bl,
           const float* __restrict__ Mb, const float* __restrict__ Sb,
           float* __restrict__ y) {
  __shared__ float outs[RT_PER_WG][16][C2_N];
  int wg = blockIdx.x;
  int b = wg / WG_PER_B;
  int rblk = wg % WG_PER_B;
  int wave = threadIdx.x >> 5;
  int lane = threadIdx.x & 31;
  int half = lane >> 4;
  int l16 = lane & 15;
  int r0 = (rblk * RT_PER_WG + wave) * 16;

  float M = Mb[b];
  float invS = 1.0f / Sb[b];

  v2f db[2];
#pragma unroll
  for (int kc = 0; kc < 2; ++kc) {
    int k = kc * 4 + half * 2;
    const float* dp = delta + (b * HW_N + r0 + l16) * C1_N + k;
    db[kc].x = dp[0];
    db[kc].y = dp[1];
  }

  v8f o0 = {0.f, 0.f, 0.f, 0.f, 0.f, 0.f, 0.f, 0.f};   // out^T c2 0..15
  v8f o1 = {0.f, 0.f, 0.f, 0.f, 0.f, 0.f, 0.f, 0.f};   // out^T c2 16..31
  const float* phib = phi + b * C1_N * HW_N;
  const float* gb = g + b * HW_N * C2_N;

  for (int mt = 0; mt < HW_N / 16; ++mt) {
    v2f a[2];
#pragma unroll
    for (int kc = 0; kc < 2; ++kc) {
      int k = kc * 4 + half * 2;
      const float* pp = phib + k * HW_N + mt * 16 + l16;
      a[kc].x = pp[0];
      a[kc].y = pp[HW_N];
    }
    v8f s = {0.f, 0.f, 0.f, 0.f, 0.f, 0.f, 0.f, 0.f};
    s = wmma4(a[0], db[0], s);
    s = wmma4(a[1], db[1], s);

    float p[8], q[8];
#pragma unroll
    for (int i = 0; i < 8; ++i) p[i] = exp2f_hw(s[i] - M);
#pragma unroll
    for (int i = 0; i < 8; ++i) q[i] = __shfl_xor(p[i], 16, 32);

    // D-layout (N=r in lanes, M=m in vgprs) -> B-layout per K-group of 4 m's.
    bool lo = (half == 0);
    v2f pb[4];
    pb[0].x = lo ? p[0] : q[2];  pb[0].y = lo ? p[1] : q[3];   // m 0..3
    pb[1].x = lo ? p[4] : q[6];  pb[1].y = lo ? p[5] : q[7];   // m 4..7
    pb[2].x = lo ? q[0] : p[2];  pb[2].y = lo ? q[1] : p[3];   // m 8..11
    pb[3].x = lo ? q[4] : p[6];  pb[3].y = lo ? q[5] : p[7];   // m 12..15

#pragma unroll
    for (int kg = 0; kg < 4; ++kg) {
      const float* gp = gb + (mt * 16 + kg * 4 + half * 2) * C2_N + l16;
      v2f ga0, ga1;
      ga0.x = gp[0];         ga0.y = gp[C2_N];        // A = g^T, c2 0..15
      ga1.x = gp[16];        ga1.y = gp[C2_N + 16];   // A = g^T, c2 16..31
      o0 = wmma4(ga0, pb[kg], o0);
      o1 = wmma4(ga1, pb[kg], o1);
    }
  }
#pragma unroll
  for (int i = 0; i < 8; ++i) { o0[i] *= invS; o1[i] *= invS; }

  // out^T tile -> LDS as outs[r_local][c2]
#pragma unroll
  for (int v = 0; v < 8; ++v) {
    outs[wave][l16][v + 8 * half] = o0[v];
    outs[wave][l16][16 + v + 8 * half] = o1[v];
  }
  __syncthreads();

  // Fused final 1x1 conv (Wl [64,32]) + bias + residual; 16 rows x 64 ch/wave.
  for (int t = 0; t < 32; ++t) {
    int rl = t >> 1;
    int co = ((t & 1) << 5) | lane;
    float acc = bl[co];
#pragma unroll
    for (int c2 = 0; c2 < C2_N; ++c2)
      acc = fmaf(Wl[co * C2_N + c2], outs[wave][rl][c2], acc);
    int oi = (b * C_N + co) * HW_N + (r0 + rl);
    y[oi] = x[oi] + acc;
  }
}

// ---------------------------------------------------------------------------
extern "C" void kernel_launch(void* const* d_in, const int* in_sizes, int n_in,
                              void* d_out, int out_size, void* d_ws, size_t ws_size,
                              hipStream_t stream) {
  const float* x  = (const float*)d_in[0];
  const float* Wd = (const float*)d_in[1];
  const float* bd = (const float*)d_in[2];
  const float* Wp = (const float*)d_in[3];
  const float* bp = (const float*)d_in[4];
  const float* Wg = (const float*)d_in[5];
  const float* bg = (const float*)d_in[6];
  const float* Wl = (const float*)d_in[7];
  const float* bl = (const float*)d_in[8];
  float* y  = (float*)d_out;
  float* ws = (float*)d_ws;

  float* delta = ws;                 // 4*4096*8   = 131072 f (log2e-scaled)
  float* phi   = ws + 131072;        // 4*8*4096   = 131072 f
  float* g     = ws + 262144;        // 4*4096*32  = 524288 f
  float* maxp  = ws + 786432;        // 128 f
  float* sump  = ws + 786560;        // 128 f
  float* Mb    = ws + 786688;        // 4 f
  float* Sb    = ws + 786692;        // 4 f   (total ~3.1 MB of ws)

  proj_kernel<<<(B_N * HW_N) / 256, 256, 0, stream>>>(x, Wd, bd, Wp, bp, Wg, bg,
                                                      delta, phi, g);
  stats_kernel<<<B_N * WG_PER_B, 256, 0, stream>>>(delta, phi, maxp, sump);
  reduce_stats_kernel<<<1, 32, 0, stream>>>(maxp, sump, Mb, Sb);
  out_kernel<<<B_N * WG_PER_B, 256, 0, stream>>>(x, delta, phi, g, Wl, bl, Mb,
                                                 Sb, y);
}